// GCN_3977139716628
// MI455X (gfx1250) — compile-verified
//
#include <hip/hip_runtime.h>

typedef __attribute__((ext_vector_type(2))) float v2f;
typedef __attribute__((ext_vector_type(8))) float v8f;

#define DIM 128

// XOR swizzle so B-operand ds_load_b64 across 16 lanes (stride 128 floats)
// hits all 64 LDS banks. Preserves 16B groups (bits 0..1 untouched).
__device__ __forceinline__ int swz(int row, int kk) {
  return row * DIM + (kk ^ ((row & 15) << 2));
}

// C[N,DIM] = A[N,DIM] @ W[DIM,DIM]^T using V_WMMA_F32_16X16X4_F32.
// One wave computes a 16-row x 128-col output stripe (8 accumulators).
__global__ __launch_bounds__(256) void gcn_gemm_xwT(
    const float* __restrict__ A, const float* __restrict__ W,
    float* __restrict__ C, int N) {
  __shared__ float lW[DIM * DIM];  // 64 KB, swizzled copy of W

  const int tid = threadIdx.x;
  // Cooperative load of W (128x128 f32) into LDS, 16 float4 per thread.
  for (int i = tid; i < DIM * (DIM / 4); i += 256) {
    const int r  = i >> 5;          // row 0..127
    const int c4 = (i & 31) << 2;   // col 0,4,...,124
    const float4 w4 = *(const float4*)(W + r * DIM + c4);
    *(float4*)&lW[swz(r, c4)] = w4;
  }
  __syncthreads();

  const int wave = tid >> 5;
  const int lane = tid & 31;
  const int half = lane >> 4;   // 0: K pair {k,k+1}, 1: {k+2,k+3}
  const int idx  = lane & 15;   // matrix row/col within tile
  const int row0 = (blockIdx.x * 8 + wave) * 16;
  if (row0 + 16 > N) return;    // wave-uniform exit; EXEC stays all-1s for WMMA

  v8f acc[8];
  const v8f vzero = {};
#pragma unroll
  for (int t = 0; t < 8; ++t) acc[t] = vzero;

  const float* arow = A + (size_t)(row0 + idx) * DIM;
  for (int k = 0; k < DIM; k += 4) {
    // A tile 16x4: lane holds 2 consecutive K values of its row.
    const v2f a = *(const v2f*)(arow + k + 2 * half);
#pragma unroll
    for (int t = 0; t < 8; ++t) {
      const int col = t * 16 + idx;
      // B tile 4x16: B[kk][col] = W[col][kk], swizzled LDS, bank-conflict free.
      const v2f b = *(const v2f*)&lW[swz(col, k + 2 * half)];
      acc[t] = __builtin_amdgcn_wmma_f32_16x16x4_f32(
          false, a, false, b, (short)0, acc[t], false, false);
    }
  }

  // C/D layout: VGPR v -> row (v + 8*half), col = t*16 + idx.
#pragma unroll
  for (int t = 0; t < 8; ++t) {
    float* crow = C + (size_t)row0 * DIM + t * 16 + idx;
#pragma unroll
    for (int v = 0; v < 8; ++v) {
      crow[(size_t)(v + 8 * half) * DIM] = acc[t][v];
    }
  }
}

// One wave32 per edge: gather S[src] (coalesced 512B), scale by norm,
// scatter-add into Z[dst] via f32 atomics (L2-resident working set).
__global__ __launch_bounds__(256) void gcn_scatter(
    const float* __restrict__ S, const float* __restrict__ norm,
    const int* __restrict__ src, const int* __restrict__ dst,
    float* __restrict__ Z, int E) {
  const int e = blockIdx.x * 8 + (threadIdx.x >> 5);
  if (e >= E) return;
  const int lane = threadIdx.x & 31;
  const int s = src[e];
  const int d = dst[e];
  const float w = norm[e];
  const float4 v = *(const float4*)(S + (size_t)s * DIM + lane * 4);
  float* zp = Z + (size_t)d * DIM + lane * 4;
  atomicAdd(zp + 0, v.x * w);
  atomicAdd(zp + 1, v.y * w);
  atomicAdd(zp + 2, v.z * w);
  atomicAdd(zp + 3, v.w * w);
}

__global__ void gcn_zero4(float4* __restrict__ p, int n4) {
  int i = blockIdx.x * blockDim.x + threadIdx.x;
  if (i < n4) p[i] = make_float4(0.f, 0.f, 0.f, 0.f);
}

__global__ void gcn_relu4(float4* __restrict__ p, int n4) {
  int i = blockIdx.x * blockDim.x + threadIdx.x;
  if (i < n4) {
    float4 v = p[i];
    v.x = fmaxf(v.x, 0.f);
    v.y = fmaxf(v.y, 0.f);
    v.z = fmaxf(v.z, 0.f);
    v.w = fmaxf(v.w, 0.f);
    p[i] = v;
  }
}

extern "C" void kernel_launch(void* const* d_in, const int* in_sizes, int n_in,
                              void* d_out, int out_size, void* d_ws, size_t ws_size,
                              hipStream_t stream) {
  const float* X   = (const float*)d_in[0];
  const float* W0  = (const float*)d_in[1];
  const float* W1  = (const float*)d_in[2];
  const float* nrm = (const float*)d_in[3];
  const int*   src = (const int*)d_in[4];
  const int*   dst = (const int*)d_in[5];

  const int N = in_sizes[0] / DIM;   // 50000
  const int E = in_sizes[3];         // 800000

  float* Z = (float*)d_out;               // output half 1: [N,128]
  float* H = Z + (size_t)N * DIM;         // output half 2: [N,128]
  float* S = (float*)d_ws;                // scratch: [N,128] (reused S0/S1)

  // Zero both output halves (scatter accumulates; d_out is poisoned).
  const int n4 = (2 * N * DIM) / 4;
  gcn_zero4<<<(n4 + 255) / 256, 256, 0, stream>>>((float4*)d_out, n4);

  const int gemmBlocks = (N / 16 + 7) / 8;  // 8 waves/block, 16 rows/wave

  // Layer 1: S0 = X @ W0^T ; H = propagate(S0) ; relu in place.
  gcn_gemm_xwT<<<gemmBlocks, 256, 0, stream>>>(X, W0, S, N);
  gcn_scatter<<<(E + 7) / 8, 256, 0, stream>>>(S, nrm, src, dst, H, E);
  const int h4 = (N * DIM) / 4;
  gcn_relu4<<<(h4 + 255) / 256, 256, 0, stream>>>((float4*)H, h4);

  // Layer 2: S1 = H @ W1^T ; Z = propagate(S1).
  gcn_gemm_xwT<<<gemmBlocks, 256, 0, stream>>>(H, W1, S, N);
  gcn_scatter<<<(E + 7) / 8, 256, 0, stream>>>(S, nrm, src, dst, Z, E);
}